// SegEncodeLoss_15960098471942
// MI455X (gfx1250) — compile-verified
//
#include <hip/hip_runtime.h>
#include <math.h>

// Problem constants (fixed by the reference: preds [16,19] f32, targets [16,1024,2048] i32)
#define BATCH        16
#define NCLS         19
#define FULLMASK     ((1u << NCLS) - 1u)          // 0x7FFFF
#define PARTS        64                           // blocks per batch
#define BLOCK        256                          // threads per block (8 wave32s)
#define ITERS        32                           // int4 (16B) loads per thread
#define PER_BLOCK_I4 (BLOCK * ITERS)              // 8192 int4 per block
#define PER_BATCH_I4 (PARTS * PER_BLOCK_I4)       // 524288 int4 = 2,097,152 labels
#define LOSSBLOCK    512
#define STAGES       4                            // async pipeline depth (3 in flight)

// ---- CDNA5 async-to-LDS path selection (probe via __has_builtin) ----
#if defined(__HIP_DEVICE_COMPILE__)
#  if __has_builtin(__builtin_amdgcn_global_load_async_to_lds_b128) && \
      __has_builtin(__builtin_amdgcn_s_wait_asynccnt)
#    define USE_ASYNC_LDS 1
#  else
#    define USE_ASYNC_LDS 0
#    warning "gfx1250 async-to-LDS builtins NOT available; falling back to direct global_load_b128"
#  endif
#else
#  define USE_ASYNC_LDS 0
#endif

#if USE_ASYNC_LDS
// Exact param types per hipcc diagnostics (probe-confirmed rounds 1-2):
//   arg0: addrspace(1) v4i32*, arg1: addrspace(3) v4i32*, then imm offset, imm cpol
typedef int v4i32 __attribute__((vector_size(16)));
typedef __attribute__((address_space(1))) v4i32* gv4i_ptr;
typedef __attribute__((address_space(3))) v4i32* lv4i_ptr;
#define ASYNC_LD(gp, lp) \
  __builtin_amdgcn_global_load_async_to_lds_b128((gv4i_ptr)(const void*)(gp), \
                                                 (lv4i_ptr)(void*)(lp), 0, 0)
#endif

// ---------------------------------------------------------------------------
// Kernel 1: per-(batch,block) class-presence bitmask over a 128 KiB slice.
// Streaming HBM-bound OR-reduction; 4-deep async-to-LDS pipeline tracked
// with ASYNCcnt (in-order completion => wait <= STAGES-1 frees oldest chunk).
// ---------------------------------------------------------------------------
__global__ __launch_bounds__(BLOCK) void presence_kernel(
    const int4* __restrict__ targets4, unsigned* __restrict__ partials) {
  const int blk = blockIdx.x;
  const int b   = blk / PARTS;
  const int p   = blk % PARTS;
  const int tid = threadIdx.x;

  const int4* src = targets4 + (size_t)b * PER_BATCH_I4 +
                    (size_t)p * PER_BLOCK_I4 + tid;

  unsigned mask = 0u;

#if USE_ASYNC_LDS
  __shared__ int4 stage[STAGES][BLOCK];   // 16 KiB, one 16B slot per lane per stage

  // Prologue: issue chunks 0..STAGES-2 (ASYNCcnt = 3)
  #pragma unroll
  for (int k = 0; k < STAGES - 1; ++k)
    ASYNC_LD(src + (size_t)k * BLOCK, &stage[k][tid]);

  #pragma unroll 8
  for (int i = 0; i < ITERS; ++i) {
    // Keep STAGES-1 chunks in flight; epilogue waits use literal immediates.
    if (i + STAGES - 1 < ITERS) {
      ASYNC_LD(src + (size_t)(i + STAGES - 1) * BLOCK, &stage[(i + STAGES - 1) % STAGES][tid]);
      __builtin_amdgcn_s_wait_asynccnt(STAGES - 1);      // chunk i complete
    } else if (i + 2 < ITERS) {
      __builtin_amdgcn_s_wait_asynccnt(2);
    } else if (i + 1 < ITERS) {
      __builtin_amdgcn_s_wait_asynccnt(1);
    } else {
      __builtin_amdgcn_s_wait_asynccnt(0);
    }
    const int4 v = stage[i % STAGES][tid];               // ds_load_b128
    mask |= (1u << v.x) | (1u << v.y) | (1u << v.z) | (1u << v.w);

    // Presence is monotone: once every lane of this wave32 has seen all 19
    // classes, further reads cannot change the result -> early exit.
    if (((i & 7) == 7) && (__ballot(mask != FULLMASK) == 0)) {
      __builtin_amdgcn_s_wait_asynccnt(0);               // drain in-flight prefetch
      break;
    }
  }
#else
  #pragma unroll 4
  for (int i = 0; i < ITERS; ++i) {
    const int4 v = src[(size_t)i * BLOCK];               // global_load_b128
    mask |= (1u << v.x) | (1u << v.y) | (1u << v.z) | (1u << v.w);
    if (((i & 7) == 7) && (__ballot(mask != FULLMASK) == 0)) break;
  }
#endif

  // Block-level OR reduction in LDS, then one partial word per block.
  __shared__ unsigned bm;
  if (tid == 0) bm = 0u;
  __syncthreads();
  atomicOr(&bm, mask);                                   // ds_or_b32
  __syncthreads();
  if (tid == 0) partials[blk] = bm;                      // no global atomics, no init needed
}

// ---------------------------------------------------------------------------
// Kernel 2: OR the 64 partials per batch, then stable BCE-with-logits mean:
//   loss(x, y) = softplus(x) - y*x,  softplus(x) = max(x,0) + log1p(exp(-|x|))
// ---------------------------------------------------------------------------
__global__ __launch_bounds__(LOSSBLOCK) void loss_kernel(
    const float* __restrict__ preds, const unsigned* __restrict__ partials,
    float* __restrict__ out) {
  __shared__ unsigned bmask[BATCH];
  __shared__ float red[LOSSBLOCK];

  const int t = threadIdx.x;
  if (t < BATCH) {
    unsigned m = 0u;
    #pragma unroll 8
    for (int p = 0; p < PARTS; ++p) m |= partials[t * PARTS + p];
    bmask[t] = m;
  }
  __syncthreads();

  float v = 0.0f;
  if (t < BATCH * NCLS) {
    const int b = t / NCLS;
    const int c = t - b * NCLS;
    const float x  = preds[t];                            // row-major [b][c] == t
    const float y  = ((bmask[b] >> c) & 1u) ? 1.0f : 0.0f;
    const float sp = fmaxf(x, 0.0f) + log1pf(__expf(-fabsf(x)));
    v = sp - y * x;
  }
  red[t] = v;
  __syncthreads();

  #pragma unroll
  for (int s = LOSSBLOCK / 2; s > 0; s >>= 1) {
    if (t < s) red[t] += red[t + s];
    __syncthreads();
  }
  if (t == 0) out[0] = red[0] * (1.0f / (float)(BATCH * NCLS));
}

// ---------------------------------------------------------------------------
extern "C" void kernel_launch(void* const* d_in, const int* in_sizes, int n_in,
                              void* d_out, int out_size, void* d_ws, size_t ws_size,
                              hipStream_t stream) {
  (void)in_sizes; (void)n_in; (void)out_size; (void)ws_size;
  const float* preds    = (const float*)d_in[0];
  const int4*  targets4 = (const int4*)d_in[1];
  unsigned*    partials = (unsigned*)d_ws;                // BATCH*PARTS = 1024 words (4 KiB)

  presence_kernel<<<BATCH * PARTS, BLOCK, 0, stream>>>(targets4, partials);
  loss_kernel<<<1, LOSSBLOCK, 0, stream>>>(preds, partials, (float*)d_out);
}